// ScaledDotProductAttention_84774064488741
// MI455X (gfx1250) — compile-verified
//
#include <hip/hip_runtime.h>

typedef __attribute__((ext_vector_type(2))) float v2f;
typedef __attribute__((ext_vector_type(8))) float v8f;
typedef int v4i_b128 __attribute__((vector_size(16)));  // matches builtin param type

#define S_LEN 2048
#define DIM   64
#define NBH   32              // B*H
#define QBLK  128             // queries per workgroup (8 waves x 16)
#define KBLK  64              // keys staged in LDS per pipeline stage
#define NT    (S_LEN / KBLK)  // 32 key tiles

// ---- CDNA5 async global->LDS path (guarded: falls back to sync copy) ----
#if defined(__AMDGCN__) && __has_builtin(__builtin_amdgcn_global_load_async_to_lds_b128)
#define HAVE_ASYNC 1
#else
#define HAVE_ASYNC 0
#endif

static __device__ __forceinline__ void wait_async_all() {
#if HAVE_ASYNC
#if __has_builtin(__builtin_amdgcn_s_wait_asynccnt)
  __builtin_amdgcn_s_wait_asynccnt(0);
#else
  asm volatile("s_wait_asynccnt 0x0" ::: "memory");
#endif
#endif
}

static __device__ __forceinline__ v8f wmma_f32(v2f a, v2f b, v8f c) {
  // D = A(16x4 f32) * B(4x16 f32) + C(16x16 f32)
  return __builtin_amdgcn_wmma_f32_16x16x4_f32(false, a, false, b, (short)0, c,
                                               false, false);
}

// max / sum reductions across the 16 lanes of each half-wave
static __device__ __forceinline__ float half16_max(float x) {
  x = fmaxf(x, __int_as_float(__builtin_amdgcn_ds_swizzle(__float_as_int(x), 0x041F)));
  x = fmaxf(x, __int_as_float(__builtin_amdgcn_ds_swizzle(__float_as_int(x), 0x081F)));
  x = fmaxf(x, __int_as_float(__builtin_amdgcn_ds_swizzle(__float_as_int(x), 0x101F)));
  x = fmaxf(x, __int_as_float(__builtin_amdgcn_ds_swizzle(__float_as_int(x), 0x201F)));
  return x;
}

static __device__ __forceinline__ float half16_sum(float x) {
  x += __int_as_float(__builtin_amdgcn_ds_swizzle(__float_as_int(x), 0x041F));
  x += __int_as_float(__builtin_amdgcn_ds_swizzle(__float_as_int(x), 0x081F));
  x += __int_as_float(__builtin_amdgcn_ds_swizzle(__float_as_int(x), 0x101F));
  x += __int_as_float(__builtin_amdgcn_ds_swizzle(__float_as_int(x), 0x201F));
  return x;
}

__global__ __launch_bounds__(256) void relattn_fwd_kernel(
    const float* __restrict__ Q, const float* __restrict__ K,
    const float* __restrict__ V, const float* __restrict__ Qp,
    const float* __restrict__ Kp, float* __restrict__ O) {
  // double-buffered key tiles + per-wave P^T scratch: 104 KB of 320 KB LDS
  __shared__ __align__(16) float Kt[2][KBLK * DIM];
  __shared__ __align__(16) float Kpt[2][KBLK * DIM];
  __shared__ __align__(16) float Vt[2][KBLK * DIM];
  __shared__ __align__(16) float Pt[8][256];

  const int head = blockIdx.x >> 4;  // 0..31  (b*H+h)
  const int qblk = blockIdx.x & 15;  // 0..15
  const int tid  = threadIdx.x;
  const int wave = tid >> 5;
  const int lane = tid & 31;
  const int lh   = lane & 15;        // row/col within 16
  const int hi   = lane >> 4;        // half-wave select (K-pair 0,1 vs 2,3)

  const size_t base = (size_t)head * S_LEN * DIM;
  const int    qrow = qblk * QBLK + wave * 16 + lh;

  const float* Kg  = K  + base;
  const float* Kpg = Kp + base;
  const float* Vg  = V  + base;

  // ---- stage key-tile t into LDS buffer `buf` (async if available) ----
  auto stage = [&](int buf, int t) {
    const float* sk  = Kg  + (size_t)t * KBLK * DIM;
    const float* skp = Kpg + (size_t)t * KBLK * DIM;
    const float* sv  = Vg  + (size_t)t * KBLK * DIM;
    float* dk  = &Kt[buf][0];
    float* dkp = &Kpt[buf][0];
    float* dv  = &Vt[buf][0];
#pragma unroll
    for (int i = 0; i < 4; ++i) {
      const int off = (tid + i * 256) * 4;  // float index, 16B per lane
#if HAVE_ASYNC
      __builtin_amdgcn_global_load_async_to_lds_b128(
          (v4i_b128*)(sk + off),  (v4i_b128*)(dk + off),  0, 0);
      __builtin_amdgcn_global_load_async_to_lds_b128(
          (v4i_b128*)(skp + off), (v4i_b128*)(dkp + off), 0, 0);
      __builtin_amdgcn_global_load_async_to_lds_b128(
          (v4i_b128*)(sv + off),  (v4i_b128*)(dv + off),  0, 0);
#else
      *(float4*)(dk + off)  = *(const float4*)(sk + off);
      *(float4*)(dkp + off) = *(const float4*)(skp + off);
      *(float4*)(dv + off)  = *(const float4*)(sv + off);
#endif
    }
  };

  // ---- Q / Q_position A-fragments, pre-scaled by 1/sqrt(64), kept in VGPRs
  v2f qa[16], qpa[16];
  {
    const float* q0 = Q  + base + (size_t)qrow * DIM;
    const float* q1 = Qp + base + (size_t)qrow * DIM;
#pragma unroll
    for (int s = 0; s < 16; ++s) {
      const int d = s * 4 + 2 * hi;
      v2f t0 = *(const v2f*)(q0 + d);
      v2f t1 = *(const v2f*)(q1 + d);
      qa[s]  = t0 * 0.125f;
      qpa[s] = t1 * 0.125f;
    }
  }

  // ---- flash-attention state
  v8f   o[4];
  float m[8], lsum[8];
#pragma unroll
  for (int dt = 0; dt < 4; ++dt)
#pragma unroll
    for (int r = 0; r < 8; ++r) o[dt][r] = 0.0f;
#pragma unroll
  for (int r = 0; r < 8; ++r) { m[r] = -1.0e30f; lsum[r] = 0.0f; }

  stage(0, 0);  // prologue: fill buffer 0

  for (int kt = 0; kt < NT; ++kt) {
    const int cur = kt & 1;
    wait_async_all();    // this wave's async fills of buf[cur] done
    __syncthreads();     // everyone's fills done / prev reads of buf[cur^1] done
    if (kt + 1 < NT) stage(cur ^ 1, kt + 1);  // overlap next tile with compute

    const float* KtC  = &Kt[cur][0];
    const float* KptC = &Kpt[cur][0];
    const float* VtC  = &Vt[cur][0];

    for (int sub = 0; sub < 4; ++sub) {  // 16-key WMMA sub-tiles
      // ---- scores: C(16q x 16k) over K-dim 128 = [Q|Qp] . [K|Kp]^T
      v8f acc;
#pragma unroll
      for (int r = 0; r < 8; ++r) acc[r] = 0.0f;
      const float* kb  = KtC  + (sub * 16 + lh) * DIM + 2 * hi;
      const float* kpb = KptC + (sub * 16 + lh) * DIM + 2 * hi;
#pragma unroll
      for (int s = 0; s < 16; ++s)
        acc = wmma_f32(qa[s], *(const v2f*)(kb + s * 4), acc);
#pragma unroll
      for (int s = 0; s < 16; ++s)
        acc = wmma_f32(qpa[s], *(const v2f*)(kpb + s * 4), acc);

      // ---- online softmax (rows q = r + 8*hi, cols = lanes within half)
      float p[8], f[8];
#pragma unroll
      for (int r = 0; r < 8; ++r) {
        const float sc = acc[r];
        const float tm = half16_max(sc);
        const float nm = fmaxf(m[r], tm);
        f[r] = __expf(m[r] - nm);
        p[r] = __expf(sc - nm);
        lsum[r] = lsum[r] * f[r] + p[r];  // per-lane partial row sum
        m[r] = nm;
      }

      // ---- P^T -> per-wave LDS scratch: Pt[key][q] (two b128 stores/lane)
      {
        float* pw = &Pt[wave][lh * 16 + 8 * hi];
        *(float4*)(pw)     = make_float4(p[0], p[1], p[2], p[3]);
        *(float4*)(pw + 4) = make_float4(p[4], p[5], p[6], p[7]);
      }
      asm volatile("s_wait_dscnt 0x0" ::: "memory");  // wave-local transpose sync

      // ---- rescale running O by exp(m_old - m_new)
#pragma unroll
      for (int dt = 0; dt < 4; ++dt)
#pragma unroll
        for (int r = 0; r < 8; ++r) o[dt][r] *= f[r];

      // ---- O += P(16x16) . V(16x64): 4 K-chunks x 4 d-tiles of f32 WMMA
      const float* pr = &Pt[wave][0];
      const float* vb = VtC + sub * 16 * DIM;
#pragma unroll
      for (int c = 0; c < 4; ++c) {
        const int k0 = c * 4 + 2 * hi;
        v2f a;
        a.x = pr[k0 * 16 + lh];
        a.y = pr[(k0 + 1) * 16 + lh];
#pragma unroll
        for (int dt = 0; dt < 4; ++dt) {
          v2f b;
          b.x = vb[k0 * DIM + dt * 16 + lh];
          b.y = vb[(k0 + 1) * DIM + dt * 16 + lh];
          o[dt] = wmma_f32(a, b, o[dt]);
        }
      }
    }
  }

  // ---- normalize and write out (row q = r + 8*hi, col d = dt*16 + lh)
  float inv[8];
#pragma unroll
  for (int r = 0; r < 8; ++r) inv[r] = 1.0f / half16_sum(lsum[r]);

  float* Og = O + base;
#pragma unroll
  for (int dt = 0; dt < 4; ++dt)
#pragma unroll
    for (int r = 0; r < 8; ++r) {
      const int q = qblk * QBLK + wave * 16 + r + 8 * hi;
      Og[(size_t)q * DIM + dt * 16 + lh] = o[dt][r] * inv[r];
    }
}

extern "C" void kernel_launch(void* const* d_in, const int* in_sizes, int n_in,
                              void* d_out, int out_size, void* d_ws,
                              size_t ws_size, hipStream_t stream) {
  (void)in_sizes; (void)n_in; (void)out_size; (void)d_ws; (void)ws_size;
  const float* Q  = (const float*)d_in[0];
  const float* K  = (const float*)d_in[1];
  const float* V  = (const float*)d_in[2];
  const float* Qp = (const float*)d_in[3];
  const float* Kp = (const float*)d_in[4];
  float* Out = (float*)d_out;

  dim3 grid(NBH * (S_LEN / QBLK));  // 32 heads * 16 q-blocks = 512 WGs
  dim3 block(256);                  // 8 wave32s
  hipLaunchKernelGGL(relattn_fwd_kernel, grid, block, 0, stream,
                     Q, K, V, Qp, Kp, Out);
}